// GATLayer_5360119185629
// MI455X (gfx1250) — compile-verified
//
#include <hip/hip_runtime.h>
#include <hip/hip_bf16.h>

typedef float v2f __attribute__((ext_vector_type(2)));
typedef float v8f __attribute__((ext_vector_type(8)));

#define NH   4
#define DIN  128
#define DOUT 32

// Monotonic float -> uint key (order-preserving) for atomicMax-based segment max.
__device__ __forceinline__ unsigned fkey(float f) {
  unsigned u = __float_as_uint(f);
  return (u >> 31) ? ~u : (u ^ 0x80000000u);
}
__device__ __forceinline__ float funkey(unsigned k) {
  unsigned u = (k >> 31) ? (k ^ 0x80000000u) : ~k;
  return __uint_as_float(u);
}

// ---------------------------------------------------------------------------
// K1: H_w[h,n,o] = sum_d H_in[n,d] * W[h,o,d] via V_WMMA_F32_16X16X4_F32.
// One wave per (head, 16-node tile). A = H_in tile (16x4), B = W slice (4x16).
// Hw stored node-major: Hw[((n*4)+h)*32 + o]  (512B contiguous per node).
// ---------------------------------------------------------------------------
__global__ __launch_bounds__(128) void k_hw_wmma(const float* __restrict__ Hin,
                                                 const float* __restrict__ W,
                                                 float* __restrict__ Hw, int N) {
  const int h     = threadIdx.x >> 5;   // wave id == head
  const int lane  = threadIdx.x & 31;
  const int m     = lane & 15;          // M (A) / N (B) index within tile
  const int khalf = lane >> 4;          // lanes 16-31 hold K+2,K+3
  const int base  = blockIdx.x * 16;

  int rowA = base + m;
  if (rowA >= N) rowA = N - 1;          // clamp loads; keep EXEC all-ones for WMMA
  const float* arow  = Hin + (size_t)rowA * DIN + 2 * khalf;
  const float* b0row = W + ((size_t)h * DOUT + m) * DIN + 2 * khalf;        // o = m
  const float* b1row = W + ((size_t)h * DOUT + m + 16) * DIN + 2 * khalf;   // o = m+16

  v8f acc0 = {0.f, 0.f, 0.f, 0.f, 0.f, 0.f, 0.f, 0.f};
  v8f acc1 = {0.f, 0.f, 0.f, 0.f, 0.f, 0.f, 0.f, 0.f};

#pragma unroll
  for (int k = 0; k < DIN; k += 4) {
    v2f a  = { arow[k],  arow[k + 1]  };
    v2f b0 = { b0row[k], b0row[k + 1] };
    v2f b1 = { b1row[k], b1row[k + 1] };
    acc0 = __builtin_amdgcn_wmma_f32_16x16x4_f32(false, a, false, b0, (short)0,
                                                 acc0, false, false);
    acc1 = __builtin_amdgcn_wmma_f32_16x16x4_f32(false, a, false, b1, (short)0,
                                                 acc1, false, false);
  }

  // D layout: VGPR g holds (M = g + 8*khalf, o = m) for acc0, o = m+16 for acc1.
#pragma unroll
  for (int g = 0; g < 8; ++g) {
    int node = base + g + 8 * khalf;
    if (node < N) {
      float* dst = Hw + ((size_t)node * NH + h) * DOUT;
      dst[m]      = acc0[g];
      dst[m + 16] = acc1[g];
    }
  }
}

// ---------------------------------------------------------------------------
// K2: per-node attention score halves: s_src[n][h], s_tgt[n][h] (interleaved
// [n][4] so the edge stage gathers one float4 per endpoint).
// ---------------------------------------------------------------------------
__global__ void k_scores(const float* __restrict__ Hw, const float* __restrict__ Asrc,
                         const float* __restrict__ Atgt, float* __restrict__ ssrc,
                         float* __restrict__ stgt, int N) {
  int n = blockIdx.x * blockDim.x + threadIdx.x;
  if (n >= N) return;
  const float* hw = Hw + (size_t)n * (NH * DOUT);
#pragma unroll
  for (int h = 0; h < NH; ++h) {
    float s1 = 0.f, s2 = 0.f;
#pragma unroll
    for (int o = 0; o < DOUT; ++o) {
      float v = hw[h * DOUT + o];
      s1 += v * Asrc[h * DOUT + o];
      s2 += v * Atgt[h * DOUT + o];
    }
    ssrc[(size_t)n * NH + h] = s1;
    stgt[(size_t)n * NH + h] = s2;
  }
}

// ---------------------------------------------------------------------------
// Init kernels
// ---------------------------------------------------------------------------
__global__ void k_init_seg(unsigned* __restrict__ segmax, float* __restrict__ segsum,
                           int n) {
  int i = blockIdx.x * blockDim.x + threadIdx.x;
  if (i < n) {
    segmax[i] = 0x007FFFFFu;  // fkey(-inf): segment_max identity
    segsum[i] = 0.f;
  }
}
__global__ void k_zero(float* __restrict__ p, int n) {
  int i = blockIdx.x * blockDim.x + threadIdx.x;
  if (i < n) p[i] = 0.f;
}

// ---------------------------------------------------------------------------
// K4: E[e][h] = leaky_relu(s_src[src][h] + s_tgt[tgt][h]); segment max by tgt.
// ---------------------------------------------------------------------------
__global__ void k_edge_logit(const int* __restrict__ src, const int* __restrict__ tgt,
                             const float* __restrict__ ssrc, const float* __restrict__ stgt,
                             float* __restrict__ Ebuf, unsigned* __restrict__ segmax,
                             int E) {
  int e = blockIdx.x * blockDim.x + threadIdx.x;
  if (e >= E) return;
  int s = src[e], t = tgt[e];
  float4 a = ((const float4*)ssrc)[s];
  float4 b = ((const float4*)stgt)[t];
  float4 v;
  v.x = a.x + b.x; v.y = a.y + b.y; v.z = a.z + b.z; v.w = a.w + b.w;
  v.x = v.x > 0.f ? v.x : 0.2f * v.x;
  v.y = v.y > 0.f ? v.y : 0.2f * v.y;
  v.z = v.z > 0.f ? v.z : 0.2f * v.z;
  v.w = v.w > 0.f ? v.w : 0.2f * v.w;
  ((float4*)Ebuf)[e] = v;
  unsigned* mrow = segmax + (size_t)t * NH;
  atomicMax(&mrow[0], fkey(v.x));
  atomicMax(&mrow[1], fkey(v.y));
  atomicMax(&mrow[2], fkey(v.z));
  atomicMax(&mrow[3], fkey(v.w));
}

// ---------------------------------------------------------------------------
// K5: E[e][h] = exp(E - seg_max[tgt][h]); segment sum by tgt.
// ---------------------------------------------------------------------------
__global__ void k_edge_exp(const int* __restrict__ tgt, float* __restrict__ Ebuf,
                           const unsigned* __restrict__ segmax,
                           float* __restrict__ segsum, int E) {
  int e = blockIdx.x * blockDim.x + threadIdx.x;
  if (e >= E) return;
  int t = tgt[e];
  float4 w = ((const float4*)Ebuf)[e];
  uint4 mk = ((const uint4*)segmax)[t];
  w.x = expf(w.x - funkey(mk.x));
  w.y = expf(w.y - funkey(mk.y));
  w.z = expf(w.z - funkey(mk.z));
  w.w = expf(w.w - funkey(mk.w));
  ((float4*)Ebuf)[e] = w;
  float* srow = segsum + (size_t)t * NH;
  atomicAdd(&srow[0], w.x);
  atomicAdd(&srow[1], w.y);
  atomicAdd(&srow[2], w.z);
  atomicAdd(&srow[3], w.w);
}

// ---------------------------------------------------------------------------
// K6: one thread per (edge, head): alpha = w / seg_sum[tgt][h];
// scatter alpha * Hw[src][h][:] into d_out[h][tgt][:].
// ---------------------------------------------------------------------------
__global__ void k_edge_scatter(const int* __restrict__ src, const int* __restrict__ tgt,
                               const float* __restrict__ Ebuf,
                               const float* __restrict__ segsum,
                               const float* __restrict__ Hw, float* __restrict__ dout,
                               int E, int N) {
  int idx = blockIdx.x * blockDim.x + threadIdx.x;
  if (idx >= E * NH) return;
  int e = idx >> 2, h = idx & 3;
  int s = src[e], t = tgt[e];
  float alpha = Ebuf[idx] / segsum[(size_t)t * NH + h];
  const float4* hw = (const float4*)(Hw + ((size_t)s * NH + h) * DOUT);
  float* out = dout + ((size_t)h * N + t) * DOUT;
#pragma unroll
  for (int q = 0; q < DOUT / 4; ++q) {
    float4 v = hw[q];
    atomicAdd(&out[4 * q + 0], alpha * v.x);
    atomicAdd(&out[4 * q + 1], alpha * v.y);
    atomicAdd(&out[4 * q + 2], alpha * v.z);
    atomicAdd(&out[4 * q + 3], alpha * v.w);
  }
}

// ---------------------------------------------------------------------------
// K7: in-place ELU (accumulation is complete; d_out is re-zeroed each call).
// ---------------------------------------------------------------------------
__global__ void k_elu(float* __restrict__ dout, int n) {
  int i = blockIdx.x * blockDim.x + threadIdx.x;
  if (i >= n) return;
  float x = dout[i];
  dout[i] = x > 0.f ? x : expm1f(x);
}

extern "C" void kernel_launch(void* const* d_in, const int* in_sizes, int n_in,
                              void* d_out, int out_size, void* d_ws, size_t ws_size,
                              hipStream_t stream) {
  const float* Hin  = (const float*)d_in[0];
  const int*   ei   = (const int*)d_in[1];
  const float* W    = (const float*)d_in[2];
  const float* Asrc = (const float*)d_in[3];
  const float* Atgt = (const float*)d_in[4];
  float* dout = (float*)d_out;

  const int N = in_sizes[0] / DIN;   // 100000
  const int E = in_sizes[1] / 2;     // 1600000
  const int* src = ei;
  const int* tgt = ei + E;

  // Workspace carve-up (all 16B-aligned): ~83 MB total.
  char* p = (char*)d_ws;
  float*    Hw     = (float*)p;    p += (size_t)N * NH * DOUT * sizeof(float);
  float*    ssrc   = (float*)p;    p += (size_t)N * NH * sizeof(float);
  float*    stgt   = (float*)p;    p += (size_t)N * NH * sizeof(float);
  unsigned* segmax = (unsigned*)p; p += (size_t)N * NH * sizeof(unsigned);
  float*    segsum = (float*)p;    p += (size_t)N * NH * sizeof(float);
  float*    Ebuf   = (float*)p;    p += (size_t)E * NH * sizeof(float);
  (void)ws_size; (void)n_in;

  const int B = 256;
  // Init accumulators.
  k_zero<<<(out_size + B - 1) / B, B, 0, stream>>>(dout, out_size);
  k_init_seg<<<(N * NH + B - 1) / B, B, 0, stream>>>(segmax, segsum, N * NH);
  // Dense stage (WMMA).
  k_hw_wmma<<<(N + 15) / 16, 128, 0, stream>>>(Hin, W, Hw, N);
  k_scores<<<(N + B - 1) / B, B, 0, stream>>>(Hw, Asrc, Atgt, ssrc, stgt, N);
  // Edge stage.
  k_edge_logit<<<(E + B - 1) / B, B, 0, stream>>>(src, tgt, ssrc, stgt, Ebuf, segmax, E);
  k_edge_exp<<<(E + B - 1) / B, B, 0, stream>>>(tgt, Ebuf, segmax, segsum, E);
  k_edge_scatter<<<(E * NH + B - 1) / B, B, 0, stream>>>(src, tgt, Ebuf, segsum, Hw,
                                                         dout, E, N);
  k_elu<<<(out_size + B - 1) / B, B, 0, stream>>>(dout, out_size);
}